// LIFSpike_17806934409272
// MI455X (gfx1250) — compile-verified
//
#include <hip/hip_runtime.h>
#include <stdint.h>

// LIF spiking-neuron scan, B=128, N=4096, T=64, float32.
// Memory-bound (AI ~0.5 flop/byte, ~268 MB traffic -> ~11.5us floor at
// 23.3 TB/s): single-pass streaming through the gfx1250 async global<->LDS
// path with padded LDS staging so the per-sequence sequential scan reads a
// private 256B row while all HBM traffic stays perfectly coalesced.

#define LIF_T 64
#define ROWS_PER_BLOCK 256
#define THREADS 256
#define ROW_BYTES (LIF_T * 4)            // 256 B per sequence
#define LDS_STRIDE (ROW_BYTES + 16)      // 272 B: 16B-aligned, skews LDS banks
#define CHUNKS ((ROWS_PER_BLOCK * ROW_BYTES) / (THREADS * 16))  // 16
#define TILE_BYTES (ROWS_PER_BLOCK * ROW_BYTES)                 // 65536
#define CHUNK_GSTRIDE (THREADS * 16)                            // 4096 B
#define CHUNK_LSTRIDE ((CHUNK_GSTRIDE / ROW_BYTES) * LDS_STRIDE) // 16*272=4352 B

typedef int v4i __attribute__((vector_size(16)));

__device__ __forceinline__ void async_g2l_b128(const void* g, const void* l) {
#if __has_builtin(__builtin_amdgcn_global_load_async_to_lds_b128)
  __builtin_amdgcn_global_load_async_to_lds_b128((v4i*)g, (v4i*)l, 0, 0);
#else
  unsigned loff = (unsigned)(uintptr_t)l;       // low 32 bits = LDS offset
  asm volatile("global_load_async_to_lds_b128 %0, %1, off"
               :: "v"(loff), "v"((uint64_t)(uintptr_t)g) : "memory");
#endif
}

__device__ __forceinline__ void async_l2g_b128(void* g, const void* l) {
#if __has_builtin(__builtin_amdgcn_global_store_async_from_lds_b128)
  __builtin_amdgcn_global_store_async_from_lds_b128((v4i*)g, (v4i*)l, 0, 0);
#else
  unsigned loff = (unsigned)(uintptr_t)l;
  asm volatile("global_store_async_from_lds_b128 %1, %0, off"
               :: "v"(loff), "v"((uint64_t)(uintptr_t)g) : "memory");
#endif
}

__device__ __forceinline__ void wait_async_zero() {
#if __has_builtin(__builtin_amdgcn_s_wait_asynccnt)
  __builtin_amdgcn_s_wait_asynccnt(0);
#else
  asm volatile("s_wait_asynccnt 0" ::: "memory");
#endif
}

__global__ __launch_bounds__(THREADS) void lif_scan_kernel(
    const float* __restrict__ x, const float* __restrict__ Vth,
    const float* __restrict__ tau, float* __restrict__ out,
    long long total_rows, int N) {
  __shared__ __align__(16) unsigned char smem[ROWS_PER_BLOCK * LDS_STRIDE];

  const int tid = threadIdx.x;
  const long long tile_row0 = (long long)blockIdx.x * ROWS_PER_BLOCK;
  const long long tile_byte0 = tile_row0 * ROW_BYTES;
  const long long total_bytes = total_rows * ROW_BYTES;
  const bool full_tile = (tile_byte0 + TILE_BYTES) <= total_bytes;

  // Per-thread bases. Flat byte f = (i*THREADS + tid)*16 within the dense
  // tile maps to LDS (f/256)*272 + (f%256); f%256 = (tid&15)*16 is
  // i-invariant, so each chunk is base + compile-time-constant stride.
  const int goff0 = tid * 16;                                   // 32-bit
  const unsigned loff0 = (unsigned)(tid >> 4) * LDS_STRIDE +
                         (unsigned)(tid & 15) * 16;
  const char* gin = (const char*)x + tile_byte0 + goff0;        // scalar+voff
  char* gout = (char*)out + tile_byte0 + goff0;

  // ---- Phase 1: coalesced async global -> LDS with 16B row padding.
  if (full_tile) {
#pragma unroll
    for (int i = 0; i < CHUNKS; ++i)
      async_g2l_b128(gin + i * CHUNK_GSTRIDE, smem + loff0 + i * CHUNK_LSTRIDE);
  } else {
#pragma unroll
    for (int i = 0; i < CHUNKS; ++i) {
      if (tile_byte0 + goff0 + (long long)i * CHUNK_GSTRIDE < total_bytes)
        async_g2l_b128(gin + i * CHUNK_GSTRIDE,
                       smem + loff0 + i * CHUNK_LSTRIDE);
    }
  }
  wait_async_zero();
  __syncthreads();

  // ---- Phase 2: each thread runs the sequential LIF scan on its own
  // padded LDS row, overwriting x_t with the spike o_t in place.
  const long long row = tile_row0 + tid;
  if (row < total_rows) {
    const int n = (int)(row % N);
    const float vth = fmaxf(Vth[n], 0.0f);                  // forward-constrain
    const float tc  = fminf(fmaxf(tau[n], 0.0f), 1.0f);     // clip(tau,0,1)
    float* rp = (float*)(smem + tid * LDS_STRIDE);
    float u = 0.0f, o = 0.0f;
#pragma unroll
    for (int t = 0; t < LIF_T; ++t) {
      const float xv = rp[t];
      u = fmaf(tc, fmaf(-vth, o, u), xv);   // tau*(u - Vth*o) + x_t
      o = (u > vth) ? 1.0f : 0.0f;          // heaviside(u - Vth)
      rp[t] = o;
    }
  }
  __syncthreads();

  // ---- Phase 3: coalesced async LDS -> global store of the spike train.
  if (full_tile) {
#pragma unroll
    for (int i = 0; i < CHUNKS; ++i)
      async_l2g_b128(gout + i * CHUNK_GSTRIDE,
                     smem + loff0 + i * CHUNK_LSTRIDE);
  } else {
#pragma unroll
    for (int i = 0; i < CHUNKS; ++i) {
      if (tile_byte0 + goff0 + (long long)i * CHUNK_GSTRIDE < total_bytes)
        async_l2g_b128(gout + i * CHUNK_GSTRIDE,
                       smem + loff0 + i * CHUNK_LSTRIDE);
    }
  }
  wait_async_zero();  // S_ENDPGM also waits-idle; explicit for clarity
}

extern "C" void kernel_launch(void* const* d_in, const int* in_sizes, int n_in,
                              void* d_out, int out_size, void* d_ws, size_t ws_size,
                              hipStream_t stream) {
  (void)n_in; (void)d_ws; (void)ws_size; (void)out_size;
  const float* x   = (const float*)d_in[0];   // [B, N, T] f32
  const float* Vth = (const float*)d_in[1];   // [N] f32
  const float* tau = (const float*)d_in[2];   // [N] f32
  float* out = (float*)d_out;                 // [B, N, T] f32

  const int N = in_sizes[1];
  const long long total_rows = (long long)in_sizes[0] / LIF_T;  // B*N
  const int blocks = (int)((total_rows + ROWS_PER_BLOCK - 1) / ROWS_PER_BLOCK);

  hipLaunchKernelGGL(lif_scan_kernel, dim3(blocks), dim3(THREADS), 0, stream,
                     x, Vth, tau, out, total_rows, N);
}